// LCL_1803886265536
// MI455X (gfx1250) — compile-verified
//
#include <hip/hip_runtime.h>
#include <stdint.h>

// ---------------------------------------------------------------------------
// Problem constants
// ---------------------------------------------------------------------------
#define N_IMG   16
#define H       1024
#define W       1024
#define N_TOTAL (16u * 1024u * 1024u)      // 16,777,216 elements
#define RANK    13421772u                  // floor(0.8 * (N_TOTAL-1)); frac == 0 exactly

#define BAND_ROWS 8                        // output rows per block
#define BANDS     (H / BAND_ROWS)          // 128
#define NBLK      (N_IMG * BANDS)          // 2048 stencil blocks

#define COARSE_BITS 20                     // histogram over bits[30:11]
#define NBINS       (1u << COARSE_BITS)    // 1,048,576 bins
#define NCHUNK      1024u                  // chunks of 1024 bins
#define REF_BINS    2048u                  // refine low 11 bits

// ---------------------------------------------------------------------------
// Workspace layout (bytes)
// ---------------------------------------------------------------------------
#define OFF_CNT       0ull                          // u32 [NBINS]  = 4 MiB
#define OFF_CHUNKCNT  (4ull << 20)                  // u32 [1024]   = 4 KiB
#define OFF_STATE     ((4ull << 20) + 4096ull)      // SelState (32 B, 8-aligned)
#define OFF_REFA      ((4ull << 20) + 4160ull)      // u32 [2048]   = 8 KiB
#define ZERO_BYTES    ((4ull << 20) + 4160ull + 8192ull)
#define ZERO_WORDS    ((unsigned)(ZERO_BYTES / 4ull))

struct SelState {
    unsigned prefixA;        // coarse bin (bits[30:11]) holding the rank (k2b)
    unsigned remRankA;       // 0-based rank within that bin               (k2b)
    unsigned cumCntBelowA;   // # elements strictly below the bin          (k2b)
    unsigned pad;
    double   sumBelow;       // sum of L with bin < prefixA   (k3, f64 atomics)
    double   totalSum;       // sum of all L                  (k3, f64 atomics)
};

// ---------------------------------------------------------------------------
// k0: zero the accumulated workspace region (ws is poisoned / reused)
// ---------------------------------------------------------------------------
__global__ void k0_zero(unsigned* __restrict__ ws, unsigned nwords) {
    unsigned stride = gridDim.x * blockDim.x;
    for (unsigned i = blockIdx.x * blockDim.x + threadIdx.x; i < nwords; i += stride)
        ws[i] = 0u;
}

// ---------------------------------------------------------------------------
// Stage a 10-row band (8 output rows + halo) into LDS via the CDNA5 async
// global->LDS path. 256 threads; thread t owns the 16 bytes [16t,16t+16) of
// every row. Out-of-image halo rows are zero-filled (conv padding=1).
// ---------------------------------------------------------------------------
__device__ __forceinline__ void stage_band_async(const float* __restrict__ pred,
                                                 float* tile /* LDS, 10*W */) {
    const int t    = threadIdx.x;            // 0..255
    const int img  = blockIdx.x >> 7;        // /128
    const int band = blockIdx.x & 127;
    const int row0 = band * BAND_ROWS;
    const float* base = pred + (size_t)img * (size_t)(H * W);

#pragma unroll
    for (int r = 0; r < BAND_ROWS + 2; ++r) {
        int gr = row0 - 1 + r;               // uniform per block -> no divergence
        float* dst = &tile[r * W + t * 4];
        if (gr >= 0 && gr < H) {
            unsigned loff = (unsigned)(uintptr_t)dst;   // low 32 bits == LDS offset
            unsigned long long ga =
                (unsigned long long)(uintptr_t)(base + (size_t)gr * W + t * 4);
            // CDNA5 async copy: global memory -> LDS, tracked by ASYNCcnt
            asm volatile("global_load_async_to_lds_b128 %0, %1, off"
                         :: "v"(loff), "v"(ga) : "memory");
        } else {
            float4 z = make_float4(0.f, 0.f, 0.f, 0.f);
            *(float4*)dst = z;
        }
    }
    asm volatile("s_wait_asynccnt 0" ::: "memory");
    __syncthreads();
}

__device__ __forceinline__ float lap_abs(const float* tile, int r, int c) {
    // LDS row index for global row g is (g - row0 + 1); output row r maps to r+1
    float up = tile[(r    ) * W + c];
    float ce = tile[(r + 1) * W + c];
    float dn = tile[(r + 2) * W + c];
    float lf = (c == 0)     ? 0.f : tile[(r + 1) * W + c - 1];
    float rt = (c == W - 1) ? 0.f : tile[(r + 1) * W + c + 1];
    return fabsf(up + dn + lf + rt - 4.0f * ce);
}

// ---------------------------------------------------------------------------
// k1: stencil + 2^20-bin count histogram (global atomics, L2-resident)
// ---------------------------------------------------------------------------
__global__ void k1_hist(const float* __restrict__ pred,
                        unsigned* __restrict__ cnt) {
    __shared__ __align__(16) float tile[(BAND_ROWS + 2) * W];
    stage_band_async(pred, tile);
    const int t = threadIdx.x;
#pragma unroll
    for (int r = 0; r < BAND_ROWS; ++r) {
#pragma unroll
        for (int j = 0; j < 4; ++j) {
            int c = t * 4 + j;
            float L = lap_abs(tile, r, c);
            unsigned bits = __float_as_uint(L);     // L >= 0 -> bits order-monotonic
            atomicAdd(&cnt[bits >> 11], 1u);        // bits[30:11], < 2^20
        }
    }
}

// ---------------------------------------------------------------------------
// k2a: reduce 1024 bins per chunk -> chunk count totals
// ---------------------------------------------------------------------------
__global__ void k2a_chunks(const unsigned* __restrict__ cnt,
                           unsigned* __restrict__ chunkCnt) {
    __shared__ unsigned sc[256];
    const int t = threadIdx.x;
    const unsigned base = blockIdx.x * 1024u;
    unsigned c = 0u;
    for (int k = t; k < 1024; k += 256) c += cnt[base + k];
    sc[t] = c; __syncthreads();
    for (int off = 128; off > 0; off >>= 1) {
        if (t < off) sc[t] += sc[t + off];
        __syncthreads();
    }
    if (t == 0) chunkCnt[blockIdx.x] = sc[0];
}

// ---------------------------------------------------------------------------
// k2b: single block, 1024 threads. Two-level inclusive scan locates the
// coarse bin holding RANK; records rank remainder + cumulative count below.
// ---------------------------------------------------------------------------
__global__ void __launch_bounds__(1024)
k2b_select(const unsigned* __restrict__ chunkCnt,
           const unsigned* __restrict__ cnt,
           SelState* __restrict__ st) {
    __shared__ unsigned sc[1024];
    __shared__ unsigned sJ, sRem, sExclCnt;
    const int t = threadIdx.x;

    // ---- chunk-level scan ----
    unsigned cc = chunkCnt[t];
    sc[t] = cc; __syncthreads();
    for (int off = 1; off < 1024; off <<= 1) {
        unsigned x = (t >= off) ? sc[t - off] : 0u; __syncthreads();
        sc[t] += x; __syncthreads();
    }
    unsigned incl = sc[t], excl = incl - cc;
    if (excl <= RANK && RANK < incl) {       // unique winner (cc > 0 there)
        sJ = (unsigned)t; sRem = RANK - excl; sExclCnt = excl;
    }
    __syncthreads();
    const unsigned j = sJ, rem = sRem, exclCnt = sExclCnt;

    // ---- bin-level scan within chunk j ----
    unsigned bc = cnt[j * 1024u + t];
    __syncthreads();
    sc[t] = bc; __syncthreads();
    for (int off = 1; off < 1024; off <<= 1) {
        unsigned x = (t >= off) ? sc[t - off] : 0u; __syncthreads();
        sc[t] += x; __syncthreads();
    }
    unsigned incl2 = sc[t], excl2 = incl2 - bc;
    if (excl2 <= rem && rem < incl2) {
        st->prefixA      = j * 1024u + (unsigned)t;
        st->remRankA     = rem - excl2;
        st->cumCntBelowA = exclCnt + excl2;
    }
}

// ---------------------------------------------------------------------------
// k3: second stencil pass (pred is L2-hot after k1). Refines the low 11 bits
// inside the rank bin AND computes totalSum / sumBelow in f64 (block-tree
// reduce + one global_atomic_add_f64 per block per quantity).
// ---------------------------------------------------------------------------
__global__ void k3_refine(const float* __restrict__ pred,
                          SelState* __restrict__ st,
                          unsigned* __restrict__ refA) {
    __shared__ __align__(16) float tile[(BAND_ROWS + 2) * W];
    __shared__ double sred[256];
    const unsigned prefixA = st->prefixA;    // uniform scalar load
    stage_band_async(pred, tile);
    const int t = threadIdx.x;

    double tsum = 0.0, bsum = 0.0;
#pragma unroll
    for (int r = 0; r < BAND_ROWS; ++r) {
#pragma unroll
        for (int j = 0; j < 4; ++j) {
            int c = t * 4 + j;
            float L = lap_abs(tile, r, c);
            unsigned bits = __float_as_uint(L);
            unsigned bin  = bits >> 11;
            tsum += (double)L;
            if (bin < prefixA) bsum += (double)L;
            if (bin == prefixA)
                atomicAdd(&refA[bits & (REF_BINS - 1u)], 1u);
        }
    }

    sred[t] = tsum; __syncthreads();
    for (int off = 128; off > 0; off >>= 1) {
        if (t < off) sred[t] += sred[t + off];
        __syncthreads();
    }
    if (t == 0) atomicAdd(&st->totalSum, sred[0]);
    __syncthreads();

    sred[t] = bsum; __syncthreads();
    for (int off = 128; off > 0; off >>= 1) {
        if (t < off) sred[t] += sred[t + off];
        __syncthreads();
    }
    if (t == 0) atomicAdd(&st->sumBelow, sred[0]);
}

// ---------------------------------------------------------------------------
// k4: single block. Locate the exact threshold value; edge/flat statistics
// follow entirely from histograms (each refined bin is one exact f32 value).
// ---------------------------------------------------------------------------
__global__ void __launch_bounds__(1024)
k4_final(const unsigned* __restrict__ refA,
         const SelState* __restrict__ st,
         float* __restrict__ out) {
    __shared__ unsigned sc[1024];
    __shared__ double   sd[1024];
    __shared__ unsigned sLow, sIncl;
    const int t = threadIdx.x;
    const unsigned prefixA = st->prefixA;
    const unsigned rem     = st->remRankA;

    unsigned c0 = refA[2 * t], c1 = refA[2 * t + 1];
    unsigned pc = c0 + c1;
    sc[t] = pc; __syncthreads();
    for (int off = 1; off < 1024; off <<= 1) {
        unsigned x = (t >= off) ? sc[t - off] : 0u; __syncthreads();
        sc[t] += x; __syncthreads();
    }
    unsigned incl = sc[t], excl = incl - pc;
    if (excl <= rem && rem < incl) {
        if (rem < excl + c0) { sLow = 2u * t;      sIncl = excl + c0; }
        else                 { sLow = 2u * t + 1u; sIncl = incl;      }
    }
    __syncthreads();
    const unsigned low = sLow, inclCnt = sIncl;

    // sum of L over elements <= threshold value within the coarse bin:
    // each refined bin s is the exact float value (prefixA<<11)|s.
    double part = 0.0;
    unsigned s0 = 2u * t, s1 = 2u * t + 1u;
    if (s0 <= low && c0)
        part += (double)c0 * (double)__uint_as_float((prefixA << 11) | s0);
    if (s1 <= low && c1)
        part += (double)c1 * (double)__uint_as_float((prefixA << 11) | s1);
    sd[t] = part; __syncthreads();
    for (int off = 512; off > 0; off >>= 1) {
        if (t < off) sd[t] += sd[t + off];
        __syncthreads();
    }

    if (t == 0) {
        // frac == 0 exactly (0.8*(n-1) = 13421772.0); and even for frac>0 the
        // mask {L > t} equals {L > v_lo} since v_lo,v_hi are adjacent order
        // statistics (no data strictly between them).
        double leCnt = (double)st->cumCntBelowA + (double)inclCnt; // # L <= v_lo
        double leSum = st->sumBelow + sd[0];                       // sum L <= v_lo
        double total = st->totalSum;
        const double n = (double)N_TOTAL;
        double nEdge   = n - leCnt;
        double edgeSum = total - leSum;
        double nFlat   = leCnt;
        double flatSum = leSum;
        float edgeMean = (nEdge > 0.0) ? (float)(edgeSum / fmax(nEdge, 1.0)) : 0.0f;
        float flatMean = (nFlat > 0.0) ? (float)(flatSum / fmax(nFlat, 1.0)) : 1e-6f;
        out[0] = flatMean / (edgeMean + 1e-6f);
    }
}

// ---------------------------------------------------------------------------
// Host launcher
// ---------------------------------------------------------------------------
extern "C" void kernel_launch(void* const* d_in, const int* in_sizes, int n_in,
                              void* d_out, int out_size, void* d_ws, size_t ws_size,
                              hipStream_t stream) {
    (void)in_sizes; (void)n_in; (void)out_size; (void)ws_size;
    const float* pred = (const float*)d_in[0];
    float* out = (float*)d_out;
    char* ws = (char*)d_ws;

    unsigned* cnt      = (unsigned*)(ws + OFF_CNT);
    unsigned* chunkCnt = (unsigned*)(ws + OFF_CHUNKCNT);
    SelState* st       = (SelState*)(ws + OFF_STATE);
    unsigned* refA     = (unsigned*)(ws + OFF_REFA);

    k0_zero   <<<2048, 256, 0, stream>>>((unsigned*)ws, ZERO_WORDS);
    k1_hist   <<<NBLK, 256, 0, stream>>>(pred, cnt);
    k2a_chunks<<<NCHUNK, 256, 0, stream>>>(cnt, chunkCnt);
    k2b_select<<<1, 1024, 0, stream>>>(chunkCnt, cnt, st);
    k3_refine <<<NBLK, 256, 0, stream>>>(pred, st, refA);
    k4_final  <<<1, 1024, 0, stream>>>(refA, st, out);
}